// GNNModel_5772436045873
// MI455X (gfx1250) — compile-verified
//
#include <hip/hip_runtime.h>
#include <hip/hip_bf16.h>

#define NN 50000
#define EE 800000
#define DD 128
#define HH 128
#define GG 500
#define H2 64
#define GPAD 512        // pooled rows padded to multiple of 64 for unpredicated WMMA
#define EPS 1e-5f

typedef __attribute__((ext_vector_type(2))) float v2f;
typedef __attribute__((ext_vector_type(8))) float v8f;

// ---------------------------------------------------------------------------
// WMMA fp32 GEMM: C[M,NC] = A[M,K] @ B[K,NC] (+ optional per-column bias).
// Compile-time K/NC -> all loads are base + immediate offset (no per-iter
// address math). Each wave owns ROWT row-tiles x one 16-col tile; one block
// has NC/16 waves covering all columns of 16*ROWT rows. M must be a multiple
// of 16*ROWT (grid.x = M / (16*ROWT)); no predication anywhere, EXEC stays
// full (WMMA requirement).
//
// Fragment layouts per CDNA5 ISA 16x16x4 f32:
//   A 16x4: lanes 0-15 rows m, K={k0,k0+1}; lanes 16-31 K={k0+2,k0+3}
//   B 4x16: lane l -> col, VGPR pair holds K={k0+2h, k0+2h+1}
//   C/D:    VGPR r: lanes 0-15 (M=r, N=l); lanes 16-31 (M=8+r, N=l)
// ---------------------------------------------------------------------------
template <int K, int NC, int ROWT, bool BIAS>
__global__ void gemm_wmma(const float* __restrict__ A,
                          const float* __restrict__ B,
                          const float* __restrict__ bias,
                          float* __restrict__ C) {
    const int lane = threadIdx.x & 31;
    const int wave = threadIdx.x >> 5;         // 0 .. NC/16-1
    const int half = lane >> 4;                // K-pair select
    const int l    = lane & 15;
    const int rowBase = blockIdx.x * (16 * ROWT);
    const int colBase = wave * 16;

    const float* Ab[ROWT];
#pragma unroll
    for (int rt = 0; rt < ROWT; ++rt)
        Ab[rt] = A + (size_t)(rowBase + rt * 16 + l) * K + half * 2;

    const float* Bc = B + colBase + l + half * 2 * NC;

    v8f acc[ROWT];
#pragma unroll
    for (int rt = 0; rt < ROWT; ++rt) acc[rt] = (v8f){};

#pragma unroll
    for (int k0 = 0; k0 < K; k0 += 4) {
        v2f b;
        b.x = Bc[k0 * NC];            // immediate offsets after unroll
        b.y = Bc[k0 * NC + NC];
#pragma unroll
        for (int rt = 0; rt < ROWT; ++rt) {
            v2f a = *(const v2f*)(Ab[rt] + k0);   // global_load_b64 + imm
            acc[rt] = __builtin_amdgcn_wmma_f32_16x16x4_f32(
                false, a, false, b, (short)0, acc[rt], false, false);
        }
    }

    const int col = colBase + l;
    float bv = 0.0f;
    if (BIAS) bv = bias[col];
#pragma unroll
    for (int rt = 0; rt < ROWT; ++rt) {
#pragma unroll
        for (int r = 0; r < 8; ++r) {
            const int row = rowBase + rt * 16 + r + half * 8;
            C[(size_t)row * NC + col] = acc[rt][r] + bv;
        }
    }
}

// ---------------------------------------------------------------------------
// Degree / symmetric normalization
// ---------------------------------------------------------------------------
__global__ void init_deg(float* deg) {
    int n = blockIdx.x * blockDim.x + threadIdx.x;
    if (n < NN) deg[n] = 1.0f;               // self loop
}
__global__ void edge_deg(const int* __restrict__ ei, float* deg) {
    int e = blockIdx.x * blockDim.x + threadIdx.x;
    if (e < EE) atomicAdd(&deg[ei[EE + e]], 1.0f);
}
__global__ void finalize_dis(float* deg) {
    int n = blockIdx.x * blockDim.x + threadIdx.x;
    if (n < NN) deg[n] = rsqrtf(fmaxf(deg[n], 1.0f));
}

// ---------------------------------------------------------------------------
// Aggregation: agg[n] = h[n]*dis[n]^2 + b (self loop + bias), then edge
// scatter-add of h[src]*dis[src]*dis[dst] into agg[dst]. One lane-quad per
// (edge, 4 features): float4 coalesced read + 4x global_atomic_add_f32.
// ---------------------------------------------------------------------------
__global__ void init_agg(const float* __restrict__ h, const float* __restrict__ dis,
                         const float* __restrict__ b, float* __restrict__ agg) {
    int t = blockIdx.x * blockDim.x + threadIdx.x;
    if (t >= NN * HH) return;
    int n = t / HH, c = t - n * HH;
    float d = dis[n];
    agg[t] = h[t] * d * d + b[c];
}

__global__ void edge_agg(const int* __restrict__ ei, const float* __restrict__ dis,
                         const float* __restrict__ h, float* __restrict__ agg) {
    int t = blockIdx.x * blockDim.x + threadIdx.x;   // t = e*32 + quad
    if (t >= EE * 32) return;
    int e = t >> 5;
    int q = (t & 31) * 4;
    int s = ei[e];
    int d = ei[EE + e];
    float nrm = dis[s] * dis[d];
    const float4 hv = *(const float4*)(h + (size_t)s * HH + q);
    float* ap = agg + (size_t)d * HH + q;
    atomicAdd(ap + 0, hv.x * nrm);
    atomicAdd(ap + 1, hv.y * nrm);
    atomicAdd(ap + 2, hv.z * nrm);
    atomicAdd(ap + 3, hv.w * nrm);
}

// ---------------------------------------------------------------------------
// BatchNorm (training mode, biased variance) via sum / sumsq. C divides 256.
// ---------------------------------------------------------------------------
__global__ void zero_f(float* p, int n) {
    int i = blockIdx.x * blockDim.x + threadIdx.x;
    if (i < n) p[i] = 0.0f;
}

__global__ void bn_stats(const float* __restrict__ x, float* __restrict__ sum,
                         float* __restrict__ sumsq, int rows, int C) {
    __shared__ float ls[256];
    __shared__ float lq[256];
    int tid = threadIdx.x;
    int c = tid % C;
    int rpb = 256 / C;
    int r0 = blockIdx.x * rpb + tid / C;
    int rs = gridDim.x * rpb;
    float s = 0.0f, q = 0.0f;
    for (int r = r0; r < rows; r += rs) {
        float v = x[(size_t)r * C + c];
        s += v; q += v * v;
    }
    ls[tid] = s; lq[tid] = q;
    __syncthreads();
    if (tid < C) {
        float S = 0.0f, Q = 0.0f;
        for (int k = tid; k < 256; k += C) { S += ls[k]; Q += lq[k]; }
        atomicAdd(&sum[tid], S);
        atomicAdd(&sumsq[tid], Q);
    }
}

__global__ void bn_relu(float* __restrict__ x, const float* __restrict__ sum,
                        const float* __restrict__ sumsq, const float* __restrict__ g,
                        const float* __restrict__ be, int rows, int C) {
    int t = blockIdx.x * blockDim.x + threadIdx.x;
    if (t >= rows * C) return;
    int c = t % C;
    float inv = 1.0f / (float)rows;
    float m = sum[c] * inv;
    float v = sumsq[c] * inv - m * m;
    float y = g[c] * (x[t] - m) * rsqrtf(v + EPS) + be[c];
    x[t] = fmaxf(y, 0.0f);
}

// ---------------------------------------------------------------------------
// Global mean pool per graph
// ---------------------------------------------------------------------------
__global__ void pool_cnt(const int* __restrict__ batch, float* cnt) {
    int n = blockIdx.x * blockDim.x + threadIdx.x;
    if (n < NN) atomicAdd(&cnt[batch[n]], 1.0f);
}
__global__ void pool_sum(const float* __restrict__ h, const int* __restrict__ batch,
                         float* __restrict__ pooled) {
    int t = blockIdx.x * blockDim.x + threadIdx.x;   // n*32 + quad
    if (t >= NN * 32) return;
    int n = t >> 5;
    int q = (t & 31) * 4;
    int gi = batch[n];
    const float4 hv = *(const float4*)(h + (size_t)n * HH + q);
    float* pp = pooled + (size_t)gi * HH + q;
    atomicAdd(pp + 0, hv.x);
    atomicAdd(pp + 1, hv.y);
    atomicAdd(pp + 2, hv.z);
    atomicAdd(pp + 3, hv.w);
}
__global__ void pool_div(float* __restrict__ pooled, const float* __restrict__ cnt) {
    int t = blockIdx.x * blockDim.x + threadIdx.x;
    if (t >= GG * HH) return;
    pooled[t] /= fmaxf(cnt[t / HH], 1.0f);
}

// ---------------------------------------------------------------------------
// Final FC2: out[g] = dot(z1[g,:64], fcW2) + fcb2 — one wave per graph
// ---------------------------------------------------------------------------
__global__ void fc2_dot(const float* __restrict__ z, const float* __restrict__ w,
                        const float* __restrict__ b, float* __restrict__ out) {
    int g = blockIdx.x;
    int lane = threadIdx.x;           // 0..31
    float s = z[(size_t)g * H2 + lane]      * w[lane]
            + z[(size_t)g * H2 + 32 + lane] * w[32 + lane];
#pragma unroll
    for (int off = 16; off > 0; off >>= 1) s += __shfl_xor(s, off, 32);
    if (lane == 0) out[g] = s + b[0];
}

// ---------------------------------------------------------------------------
// Launch
// ---------------------------------------------------------------------------
static inline int cdiv(int a, int b) { return (a + b - 1) / b; }

extern "C" void kernel_launch(void* const* d_in, const int* in_sizes, int n_in,
                              void* d_out, int out_size, void* d_ws, size_t ws_size,
                              hipStream_t stream) {
    const float* x    = (const float*)d_in[0];
    const int*   ei   = (const int*)  d_in[1];
    const int*   batch= (const int*)  d_in[2];
    const float* W[3]  = {(const float*)d_in[3],  (const float*)d_in[7],  (const float*)d_in[11]};
    const float* bb[3] = {(const float*)d_in[4],  (const float*)d_in[8],  (const float*)d_in[12]};
    const float* gg[3] = {(const float*)d_in[5],  (const float*)d_in[9],  (const float*)d_in[13]};
    const float* bg[3] = {(const float*)d_in[6],  (const float*)d_in[10], (const float*)d_in[14]};
    const float* fcW1 = (const float*)d_in[15];
    const float* fcb1 = (const float*)d_in[16];
    const float* fcg1 = (const float*)d_in[17];
    const float* fcbe1= (const float*)d_in[18];
    const float* fcW2 = (const float*)d_in[19];
    const float* fcb2 = (const float*)d_in[20];
    float* out = (float*)d_out;

    float* ws     = (float*)d_ws;
    float* dis    = ws;                          // NN
    float* hbuf   = dis + NN;                    // NN*HH
    float* agg    = hbuf + (size_t)NN * HH;      // NN*HH
    float* stats  = agg  + (size_t)NN * HH;      // 256 (sum128 + sumsq128)
    float* pooled = stats + 256;                 // GPAD*HH (rows >= GG stay zero)
    float* cnt    = pooled + (size_t)GPAD * HH;  // GG
    float* z1     = cnt + GG;                    // GPAD*H2 (rows >= GG unused)
    float* stats2 = z1 + (size_t)GPAD * H2;      // 128 (sum64 + sumsq64)

    const int T = 256;

    // --- degree / symmetric normalization ---
    init_deg    <<<cdiv(NN, T), T, 0, stream>>>(dis);
    edge_deg    <<<cdiv(EE, T), T, 0, stream>>>(ei, dis);
    finalize_dis<<<cdiv(NN, T), T, 0, stream>>>(dis);

    // --- 3x (GCN conv -> BN -> ReLU) ---
    // 50000 rows = 625 blocks x (5 row-tiles x 16); 8 waves cover 128 cols.
    const float* layer_in = x;
    for (int L = 0; L < 3; ++L) {
        gemm_wmma<HH, HH, 5, false><<<NN / 80, 256, 0, stream>>>(layer_in, W[L], nullptr, hbuf);
        init_agg <<<cdiv(NN * HH, T), T, 0, stream>>>(hbuf, dis, bb[L], agg);
        edge_agg <<<cdiv(EE * 32, T), T, 0, stream>>>(ei, dis, hbuf, agg);
        zero_f   <<<1, 256, 0, stream>>>(stats, 256);
        bn_stats <<<512, 256, 0, stream>>>(agg, stats, stats + 128, NN, HH);
        bn_relu  <<<cdiv(NN * HH, T), T, 0, stream>>>(agg, stats, stats + 128, gg[L], bg[L], NN, HH);
        layer_in = agg;
    }

    // --- global mean pool (pooled padded to GPAD rows, pad stays zero) ---
    zero_f  <<<cdiv(GPAD * HH + GG, T), T, 0, stream>>>(pooled, GPAD * HH + GG);
    pool_cnt<<<cdiv(NN, T), T, 0, stream>>>(batch, cnt);
    pool_sum<<<cdiv(NN * 32, T), T, 0, stream>>>(agg, batch, pooled);
    pool_div<<<cdiv(GG * HH, T), T, 0, stream>>>(pooled, cnt);

    // --- FC1 (+bias) via WMMA over GPAD rows: 8 blocks x (4 row-tiles x 16),
    //     4 waves cover 64 cols. Rows >= GG produce unused bias rows in z1.
    gemm_wmma<HH, H2, 4, true><<<GPAD / 64, 128, 0, stream>>>(pooled, fcW1, fcb1, z1);
    zero_f  <<<1, 128, 0, stream>>>(stats2, 128);
    bn_stats<<<64, 256, 0, stream>>>(z1, stats2, stats2 + 64, GG, H2);
    bn_relu <<<cdiv(GG * H2, T), T, 0, stream>>>(z1, stats2, stats2 + 64, fcg1, fcbe1, GG, H2);

    // --- FC2 ---
    fc2_dot<<<GG, 32, 0, stream>>>(z1, fcW2, fcb2, out);
}